// GraphEncoder_70858370450219
// MI455X (gfx1250) — compile-verified
//
#include <hip/hip_runtime.h>
#include <hip/hip_bf16.h>
#include <math.h>

typedef float v2f __attribute__((ext_vector_type(2)));
typedef float v8f __attribute__((ext_vector_type(8)));

#define DIM_IN  128
#define DIM_HID 256
#define DIM_OUT 128

// ---------------------------------------------------------------- utilities
__device__ __forceinline__ void atomicMaxF(float* addr, float val) {
    // order-preserving bit tricks: ints compare like floats for >=0,
    // unsigned-min compares correctly for negatives.
    if (val >= 0.0f) atomicMax((int*)addr, __float_as_int(val));
    else             atomicMin((unsigned int*)addr, __float_as_uint(val));
}

// ---------------------------------------------------------------- degree / dinv
__global__ void k_init_deg(float* deg, int n) {
    int i = blockIdx.x * blockDim.x + threadIdx.x;
    if (i < n) deg[i] = 1.0f;                      // self-loop contribution
}

__global__ void k_deg_edges(const int* __restrict__ cols, float* deg, int e) {
    int i = blockIdx.x * blockDim.x + threadIdx.x;
    if (i < e) atomicAdd(&deg[cols[i]], 1.0f);
}

__global__ void k_dinv(float* deg, int n) {
    int i = blockIdx.x * blockDim.x + threadIdx.x;
    if (i < n) {
        float d = deg[i];
        deg[i] = (d > 0.0f) ? rsqrtf(d) : 0.0f;
    }
}

// ---------------------------------------------------------------- fp32 WMMA GEMM
// C[M x Nc] = (RELU ? relu(A) : A)[M x K] @ W[K x Nc]
// One wave computes a 16 x (16*NT) output strip: the A fragment is loaded once
// per K-step and reused across NT back-to-back V_WMMA_F32_16X16X4_F32 issues.
template <int NT, bool RELU>
__global__ void k_gemm_wmma_f32(const float* __restrict__ A,
                                const float* __restrict__ W,
                                float* __restrict__ C,
                                int M, int Nc, int K) {
    int wave = (blockIdx.x * blockDim.x + threadIdx.x) >> 5;
    int lane = threadIdx.x & 31;
    int nStripsN = Nc / (16 * NT);
    int nTilesM  = M >> 4;                     // M divisible by 16 here
    if (wave >= nTilesM * nStripsN) return;    // wave-uniform branch
    int strip = wave % nStripsN;
    int tileM = wave / nStripsN;

    int lo = lane & 15;
    int hi = lane >> 4;                        // 0 or 1

    const float* arow  = A + (size_t)(tileM * 16 + lo) * K;
    const float* wbase = W + strip * 16 * NT + lo;

    v8f acc[NT] = {};
    for (int k0 = 0; k0 < K; k0 += 4) {
        int ka = k0 + 2 * hi;
        // A fragment: lanes 0-15 -> K = k0,k0+1 ; lanes 16-31 -> K = k0+2,k0+3
        float2 av = *(const float2*)(arow + ka);       // 8B aligned (K mult of 4)
        v2f a; a.x = av.x; a.y = av.y;
        if (RELU) { a.x = fmaxf(a.x, 0.0f); a.y = fmaxf(a.y, 0.0f); }

        const float* wr0 = wbase + (size_t)ka * Nc;
        const float* wr1 = wbase + (size_t)(ka + 1) * Nc;
        #pragma unroll
        for (int t = 0; t < NT; ++t) {
            // B fragment: VGPR0 row K=ka, VGPR1 row K=ka+1
            v2f b; b.x = wr0[t * 16]; b.y = wr1[t * 16];
            acc[t] = __builtin_amdgcn_wmma_f32_16x16x4_f32(
                         /*neg_a=*/false, a, /*neg_b=*/false, b,
                         /*c_mod=*/(short)0, acc[t],
                         /*reuse_a=*/false, /*reuse_b=*/false);
        }
    }

    // C/D layout: VGPR r -> row M = r (lanes 0-15), M = r+8 (lanes 16-31)
    #pragma unroll
    for (int t = 0; t < NT; ++t) {
        float* crow = C + (size_t)(tileM * 16 + hi * 8) * Nc
                        + strip * 16 * NT + t * 16 + lo;
        #pragma unroll
        for (int r = 0; r < 8; ++r)
            crow[(size_t)r * Nc] = acc[t][r];
    }
}

// ---------------------------------------------------------------- aggregation
// out[n][f] = xw[n][f] * dinv[n]^2 + bias[f]   (self-loop term + bias)
__global__ void k_agg_init(const float* __restrict__ xw,
                           const float* __restrict__ dinv,
                           const float* __restrict__ bias,
                           float* __restrict__ out,
                           int n, int F, int logF) {
    int idx = blockIdx.x * blockDim.x + threadIdx.x;
    if (idx >= n * F) return;
    int node = idx >> logF;
    int f = idx & (F - 1);
    float d = dinv[node];
    out[idx] = xw[idx] * d * d + bias[f];
}

// out[c][f] += xw[r][f] * dinv[r]*dinv[c]   (edge scatter, float4 per thread)
__global__ void k_agg_edges(const float* __restrict__ xw,
                            const int* __restrict__ rows,
                            const int* __restrict__ cols,
                            const float* __restrict__ dinv,
                            float* __restrict__ out,
                            int e, int F, int logF4) {
    int idx = blockIdx.x * blockDim.x + threadIdx.x;
    int ee = idx >> logF4;
    if (ee >= e) return;
    int f4 = idx & ((1 << logF4) - 1);
    int r = rows[ee];
    int c = cols[ee];
    float nrm = dinv[r] * dinv[c];
    float4 v = ((const float4*)(xw + (size_t)r * F))[f4];
    float* o = out + (size_t)c * F + (f4 << 2);
    atomicAdd(o + 0, v.x * nrm);
    atomicAdd(o + 1, v.y * nrm);
    atomicAdd(o + 2, v.z * nrm);
    atomicAdd(o + 3, v.w * nrm);
}

// ---------------------------------------------------------------- pooling
__global__ void k_pool_init(float* mean_acc, float* counts,
                            float* zg /* d_out + N*DIM_OUT */, int b) {
    int i = blockIdx.x * blockDim.x + threadIdx.x;
    if (i < b * DIM_OUT) {
        mean_acc[i] = 0.0f;
        int bb = i >> 7, f = i & (DIM_OUT - 1);
        zg[(size_t)bb * (2 * DIM_OUT) + DIM_OUT + f] = -INFINITY; // segment_max identity
    }
    if (i < b) counts[i] = 0.0f;
}

__global__ void k_counts(const int* __restrict__ batch, float* counts, int n) {
    int i = blockIdx.x * blockDim.x + threadIdx.x;
    if (i < n) atomicAdd(&counts[batch[i]], 1.0f);
}

__global__ void k_pool_acc(const float* __restrict__ z,
                           const int* __restrict__ batch,
                           float* mean_acc, float* zg, int n) {
    int idx = blockIdx.x * blockDim.x + threadIdx.x;
    if (idx >= n * DIM_OUT) return;
    int node = idx >> 7;
    int f = idx & (DIM_OUT - 1);
    int b = batch[node];
    float v = z[idx];
    atomicAdd(&mean_acc[(size_t)b * DIM_OUT + f], v);
    atomicMaxF(&zg[(size_t)b * (2 * DIM_OUT) + DIM_OUT + f], v);
}

__global__ void k_pool_final(const float* __restrict__ mean_acc,
                             const float* __restrict__ counts,
                             float* zg, int b) {
    int idx = blockIdx.x * blockDim.x + threadIdx.x;
    if (idx >= b * DIM_OUT) return;
    int bb = idx >> 7;
    int f = idx & (DIM_OUT - 1);
    zg[(size_t)bb * (2 * DIM_OUT) + f] = mean_acc[idx] / fmaxf(counts[bb], 1.0f);
}

// ---------------------------------------------------------------- launch
extern "C" void kernel_launch(void* const* d_in, const int* in_sizes, int n_in,
                              void* d_out, int out_size, void* d_ws, size_t ws_size,
                              hipStream_t stream) {
    const float* x    = (const float*)d_in[0];
    const int*   ei   = (const int*)d_in[1];     // [2, E] row-major
    const int*   batch= (const int*)d_in[2];
    const float* W1   = (const float*)d_in[3];
    const float* b1   = (const float*)d_in[4];
    const float* W2   = (const float*)d_in[5];
    const float* b2   = (const float*)d_in[6];

    const int N = in_sizes[0] / DIM_IN;
    const int E = in_sizes[1] / 2;
    const int B = (out_size - N * DIM_OUT) / (2 * DIM_OUT);

    const int* rows = ei;
    const int* cols = ei + E;

    float* ws       = (float*)d_ws;
    float* buf1     = ws;                          // N*256 (xw1, later xw2)
    float* buf2     = buf1 + (size_t)N * DIM_HID;  // N*256 (layer-1 pre-relu acts)
    float* dinv     = buf2 + (size_t)N * DIM_HID;  // N (deg -> dinv in place)
    float* counts   = dinv + N;                    // B
    float* mean_acc = counts + B;                  // B*128

    float* z  = (float*)d_out;                     // [N,128]
    float* zg = z + (size_t)N * DIM_OUT;           // [B,256] = [mean | max]

    const int T = 256;

    // ---- degrees
    k_init_deg<<<(N + T - 1) / T, T, 0, stream>>>(dinv, N);
    k_deg_edges<<<(E + T - 1) / T, T, 0, stream>>>(cols, dinv, E);
    k_dinv<<<(N + T - 1) / T, T, 0, stream>>>(dinv, N);

    // ---- layer 1: xw1 = x @ W1  (fp32 WMMA, 16x64 strip per wave)
    {
        int waves = (N / 16) * (DIM_HID / 64);
        k_gemm_wmma_f32<4, false><<<(waves + 7) / 8, T, 0, stream>>>(
            x, W1, buf1, N, DIM_HID, DIM_IN);
    }
    // aggregate: buf2 = scatter(xw1 * norm) + self-loop + b1
    k_agg_init<<<((size_t)N * DIM_HID + T - 1) / T, T, 0, stream>>>(
        buf1, dinv, b1, buf2, N, DIM_HID, 8);
    {
        long long tot = (long long)E * (DIM_HID / 4);
        k_agg_edges<<<(tot + T - 1) / T, T, 0, stream>>>(
            buf1, rows, cols, dinv, buf2, E, DIM_HID, 6);   // 64 = 2^6 float4/edge
    }

    // ---- layer 2: xw2 = relu(buf2) @ W2  (relu fused into A-load, compile-time)
    {
        int waves = (N / 16) * (DIM_OUT / 64);
        k_gemm_wmma_f32<4, true><<<(waves + 7) / 8, T, 0, stream>>>(
            buf2, W2, buf1, N, DIM_OUT, DIM_HID);
    }
    // aggregate straight into d_out z region
    k_agg_init<<<((size_t)N * DIM_OUT + T - 1) / T, T, 0, stream>>>(
        buf1, dinv, b2, z, N, DIM_OUT, 7);
    {
        long long tot = (long long)E * (DIM_OUT / 4);
        k_agg_edges<<<(tot + T - 1) / T, T, 0, stream>>>(
            buf1, rows, cols, dinv, z, E, DIM_OUT, 5);      // 32 = 2^5 float4/edge
    }

    // ---- pooling
    k_pool_init<<<(B * DIM_OUT + T - 1) / T, T, 0, stream>>>(mean_acc, counts, zg, B);
    k_counts<<<(N + T - 1) / T, T, 0, stream>>>(batch, counts, N);
    k_pool_acc<<<((size_t)N * DIM_OUT + T - 1) / T, T, 0, stream>>>(
        z, batch, mean_acc, zg, N);
    k_pool_final<<<(B * DIM_OUT + T - 1) / T, T, 0, stream>>>(mean_acc, counts, zg, B);
}